// OurModel_76201309765676
// MI455X (gfx1250) — compile-verified
//
#include <hip/hip_runtime.h>

typedef float v2f __attribute__((ext_vector_type(2)));
typedef float v8f __attribute__((ext_vector_type(8)));

#define EMB 64
#define BATCH_SZ 4096

// ---------------------------------------------------------------------------
// rows[] is sorted -> CSR row pointers by binary search (lower_bound).
// ---------------------------------------------------------------------------
__global__ void build_row_ptr_kernel(const int* __restrict__ rows, int nnz,
                                     int* __restrict__ rp, int n) {
    int r = blockIdx.x * blockDim.x + threadIdx.x;
    if (r > n) return;
    int lo = 0, hi = nnz;
    while (lo < hi) {
        int mid = (lo + hi) >> 1;
        if (rows[mid] < r) lo = mid + 1; else hi = mid;
    }
    rp[r] = lo;
}

// ---------------------------------------------------------------------------
// E0 = concat(user_emb, item_emb), float4 (global_load/store_b128).
// ---------------------------------------------------------------------------
__global__ void concat_copy_kernel(const float4* __restrict__ u, long un4,
                                   const float4* __restrict__ it, long in4,
                                   float4* __restrict__ dst) {
    long i = (long)blockIdx.x * blockDim.x + threadIdx.x;
    if (i >= un4 + in4) return;
    dst[i] = (i < un4) ? u[i] : it[i - un4];
}

// ---------------------------------------------------------------------------
// Windowed CSR SpMM on V_WMMA_F32_16X16X4_F32 (wave32).
// One wave owns 16 consecutive output rows; its nnz span [p0,p1) is
// contiguous (rows sorted). Per chunk of 4 nnz:
//   A[16x4]  : vals scattered to local row    (M = lane&15, K = 2*half + v)
//   B[4x16]  : gathered src rows, 16-dim tile (K = 2*half + v, N = lane&15)
//   C[16x16] += A*B  -> 4 dim-tiles cover EMB=64.
// Main loop is clamp-free (full chunks only); one clamped tail chunk handles
// the remainder with A=0 in padded K slots. Stores are exclusive per window
// -> no atomics, no zero-init, deterministic. Requires n % 16 == 0.
// ---------------------------------------------------------------------------
__global__ __launch_bounds__(256) void spmm_wmma_kernel(
    const int* __restrict__ rp, const int* __restrict__ rows,
    const int* __restrict__ cols, const float* __restrict__ vals,
    const float* __restrict__ src, float* __restrict__ dst, int n) {
    const int lane = threadIdx.x & 31;
    const int wave = threadIdx.x >> 5;
    const int window = blockIdx.x * 8 + wave;
    const int row0 = window * 16;
    if (row0 >= n) return;                        // wave-uniform exit

    const int kh  = lane >> 4;                    // half-wave id
    const int l16 = lane & 15;
    const int myrow = row0 + l16;

    const int p0 = __builtin_amdgcn_readfirstlane(rp[row0]);
    const int p1 = __builtin_amdgcn_readfirstlane(rp[row0 + 16]);
    const int span = p1 - p0;
    const int pend = p0 + (span & ~3);            // end of full chunks

    v8f c0 = {}, c1 = {}, c2 = {}, c3 = {};

    // per-half base pointers into the index streams; bumped by 4/chunk
    const int*   rptr = rows + p0 + kh * 2;
    const int*   cptr = cols + p0 + kh * 2;
    const float* vptr = vals + p0 + kh * 2;
    const float* sbase = src + l16;

    for (int p = p0; p < pend; p += 4) {
        __builtin_prefetch(cptr + 64, 0, 1);
        __builtin_prefetch(vptr + 64, 0, 1);

        const int   r0 = rptr[0], r1 = rptr[1];
        const int   q0 = cptr[0], q1 = cptr[1];
        const float w0 = vptr[0], w1 = vptr[1];
        rptr += 4; cptr += 4; vptr += 4;

        v2f a;
        a.x = (r0 == myrow) ? w0 : 0.0f;
        a.y = (r1 == myrow) ? w1 : 0.0f;

        const float* s0 = sbase + (long)q0 * EMB;
        const float* s1 = sbase + (long)q1 * EMB;

        v2f b;
        b.x = s0[0];  b.y = s1[0];
        c0 = __builtin_amdgcn_wmma_f32_16x16x4_f32(false, a, false, b, (short)0, c0, false, false);
        b.x = s0[16]; b.y = s1[16];
        c1 = __builtin_amdgcn_wmma_f32_16x16x4_f32(false, a, false, b, (short)0, c1, false, false);
        b.x = s0[32]; b.y = s1[32];
        c2 = __builtin_amdgcn_wmma_f32_16x16x4_f32(false, a, false, b, (short)0, c2, false, false);
        b.x = s0[48]; b.y = s1[48];
        c3 = __builtin_amdgcn_wmma_f32_16x16x4_f32(false, a, false, b, (short)0, c3, false, false);
    }

    if (pend < p1) {                              // single clamped tail chunk
        const int i0 = pend + kh * 2;
        const int i1 = i0 + 1;
        const int last = p1 - 1;
        const int i0c = i0 < last ? i0 : last;
        const int i1c = i1 < last ? i1 : last;

        const int   r0 = rows[i0c], r1 = rows[i1c];
        const int   q0 = cols[i0c], q1 = cols[i1c];
        const float w0 = vals[i0c], w1 = vals[i1c];

        v2f a;
        a.x = (i0 < p1 && r0 == myrow) ? w0 : 0.0f;
        a.y = (i1 < p1 && r1 == myrow) ? w1 : 0.0f;

        const float* s0 = sbase + (long)q0 * EMB;
        const float* s1 = sbase + (long)q1 * EMB;

        v2f b;
        b.x = s0[0];  b.y = s1[0];
        c0 = __builtin_amdgcn_wmma_f32_16x16x4_f32(false, a, false, b, (short)0, c0, false, false);
        b.x = s0[16]; b.y = s1[16];
        c1 = __builtin_amdgcn_wmma_f32_16x16x4_f32(false, a, false, b, (short)0, c1, false, false);
        b.x = s0[32]; b.y = s1[32];
        c2 = __builtin_amdgcn_wmma_f32_16x16x4_f32(false, a, false, b, (short)0, c2, false, false);
        b.x = s0[48]; b.y = s1[48];
        c3 = __builtin_amdgcn_wmma_f32_16x16x4_f32(false, a, false, b, (short)0, c3, false, false);
    }

    // C/D layout: VGPR r -> M = r + 8*half, N = l16.
    float* dbase = dst + ((long)row0 + kh * 8) * EMB + l16;
#pragma unroll
    for (int r = 0; r < 8; ++r) {
        dbase[(long)r * EMB + 0]  = c0[r];
        dbase[(long)r * EMB + 16] = c1[r];
        dbase[(long)r * EMB + 32] = c2[r];
        dbase[(long)r * EMB + 48] = c3[r];
    }
}

// ---------------------------------------------------------------------------
// out_slab (+)= 0.25 * E[idx[b] + off], folded mean -> no 3rd 38MB buffer.
// ---------------------------------------------------------------------------
__global__ void gather_acc_kernel(const float* __restrict__ E,
                                  const int* __restrict__ idx, int off,
                                  float* __restrict__ out, float scale, int init) {
    int t = blockIdx.x * blockDim.x + threadIdx.x;
    if (t >= BATCH_SZ * (EMB / 4)) return;
    int b = t >> 4, q = t & 15;
    long row = (long)(idx[b] + off);
    const float4* sp = (const float4*)(E + row * EMB);
    float4 v = sp[q];
    float4* op = (float4*)(out + (long)b * EMB) + q;
    float4 o;
    if (init) {
        o.x = scale * v.x; o.y = scale * v.y; o.z = scale * v.z; o.w = scale * v.w;
    } else {
        float4 cur = *op;
        o.x = cur.x + scale * v.x; o.y = cur.y + scale * v.y;
        o.z = cur.z + scale * v.z; o.w = cur.w + scale * v.w;
    }
    *op = o;
}

// ---------------------------------------------------------------------------
extern "C" void kernel_launch(void* const* d_in, const int* in_sizes, int n_in,
                              void* d_out, int out_size, void* d_ws, size_t ws_size,
                              hipStream_t stream) {
    const int*   rows1  = (const int*)d_in[0];
    const int*   cols1  = (const int*)d_in[1];
    const float* vals1  = (const float*)d_in[2];
    const int*   rows2  = (const int*)d_in[3];
    const int*   cols2  = (const int*)d_in[4];
    const float* vals2  = (const float*)d_in[5];
    const float* ue1    = (const float*)d_in[6];
    const float* ie1    = (const float*)d_in[7];
    const float* ue2    = (const float*)d_in[8];
    const float* ie2    = (const float*)d_in[9];
    const int*   users1 = (const int*)d_in[10];
    const int*   pos1   = (const int*)d_in[11];
    const int*   neg1   = (const int*)d_in[12];
    const int*   users2 = (const int*)d_in[13];
    const int*   pos2   = (const int*)d_in[14];
    const int*   neg2   = (const int*)d_in[15];
    float* out = (float*)d_out;

    const int NU1 = 100000, NI1 = 50000;
    const int NU2 = 100000, NI2 = 50000;
    const int N1 = NU1 + NI1, N2 = NU2 + NI2;
    const int NMAX = (N1 > N2) ? N1 : N2;
    const float scale = 0.25f;   // mean over 4 embedding states

    float* Ea = (float*)d_ws;
    float* Eb = Ea + (size_t)NMAX * EMB;
    int*   rp = (int*)(Eb + (size_t)NMAX * EMB);

    struct G {
        const int *rows, *cols; const float *vals;
        const float *ue, *ie; int nu, ni, n, nnz;
        const int* gidx[3]; int goff[3]; int slab[3];
    };
    G gs[2] = {
        { rows1, cols1, vals1, ue1, ie1, NU1, NI1, N1, in_sizes[0],
          { users1, pos1, neg1 }, { 0, NU1, NU1 }, { 0, 1, 2 } },
        { rows2, cols2, vals2, ue2, ie2, NU2, NI2, N2, in_sizes[3],
          { users2, pos2, neg2 }, { 0, NU2, NU2 }, { 3, 4, 5 } },
    };

    const int gthreads = BATCH_SZ * (EMB / 4);
    const int gblocks  = (gthreads + 255) / 256;

    for (int gi = 0; gi < 2; ++gi) {
        G& g = gs[gi];

        build_row_ptr_kernel<<<(g.n + 1 + 255) / 256, 256, 0, stream>>>(
            g.rows, g.nnz, rp, g.n);

        long total4 = (long)g.n * EMB / 4;
        concat_copy_kernel<<<(int)((total4 + 255) / 256), 256, 0, stream>>>(
            (const float4*)g.ue, (long)g.nu * EMB / 4,
            (const float4*)g.ie, (long)g.ni * EMB / 4, (float4*)Ea);

        for (int s = 0; s < 3; ++s)
            gather_acc_kernel<<<gblocks, 256, 0, stream>>>(
                Ea, g.gidx[s], g.goff[s],
                out + (long)g.slab[s] * BATCH_SZ * EMB, scale, 1);

        float* cur = Ea; float* nxt = Eb;
        const int windows = g.n / 16;           // 150000 % 16 == 0
        const int sblocks = (windows + 7) / 8;  // 8 waves per 256-thread block
        for (int l = 0; l < 3; ++l) {
            spmm_wmma_kernel<<<sblocks, 256, 0, stream>>>(
                rp, g.rows, g.cols, g.vals, cur, nxt, g.n);
            for (int s = 0; s < 3; ++s)
                gather_acc_kernel<<<gblocks, 256, 0, stream>>>(
                    nxt, g.gidx[s], g.goff[s],
                    out + (long)g.slab[s] * BATCH_SZ * EMB, scale, 0);
            float* tmp = cur; cur = nxt; nxt = tmp;
        }
    }
}